// PConv3D_25821343383765
// MI455X (gfx1250) — compile-verified
//
#include <hip/hip_runtime.h>

typedef float v2f __attribute__((ext_vector_type(2)));
typedef float v8f __attribute__((ext_vector_type(8)));

#define PNT   8192   // points
#define KNB   32     // neighbors
#define FBD   45     // NF*KB
#define FBP   48     // fb padded to 3 WMMA tiles
#define PT    8      // points per block
#define UREG  (80*FBP + 8)   // per-point U region (floats), small pad vs banks

struct Smem {
  float t1r[32][80];     // T1 = dc[nbr]*pha  (re)
  float t1i[32][80];     //                   (im)
  float gpr[32][48];     // GP = G*phb        (re), cols 45..47 zero-padded
  float gpi[32][48];     //                   (im)
  float ur[PT][UREG];    // U (re) per point, row stride FBP
  float ui[PT][UREG];    // U (im)
};

__device__ __forceinline__ v8f wmma_f32(v2f a, v2f b, v8f c) {
  // D = A(16x4) * B(4x16) + C, fp32, wave32
  return __builtin_amdgcn_wmma_f32_16x16x4_f32(false, a, false, b, (short)0, c,
                                               false, false);
}

__global__ __launch_bounds__(256) void PConv3D_kernel(
    const float* __restrict__ data,   // (P,16,5,2)
    const float* __restrict__ pos,    // (P,3)
    const int*   __restrict__ nbrs,   // (P,32)
    const float* __restrict__ wts,    // (5,16,5,10,16,2)
    const float* __restrict__ bias,   // (16)
    float* __restrict__ out)          // (P,16,5,2)
{
  extern __shared__ __align__(16) char smem_raw[];
  Smem& s = *reinterpret_cast<Smem*>(smem_raw);

  const int tid  = threadIdx.x;
  const int lane = tid & 31;
  const int wave = tid >> 5;       // 0..7
  const int kh   = lane >> 4;      // lane half (K split in WMMA fragments)
  const int lc   = lane & 15;      // fragment row/col lane index
  const int pg0  = blockIdx.x * PT;

  // ================= per-point: stage T1/GP, then GEMM1 -> U =================
  for (int pl = 0; pl < PT; ++pl) {
    const int p = pg0 + pl;
    __syncthreads();   // staging buffers free from previous point's GEMM1

    { // ---- stage: thread = (neighbor n, column group g) ----
      const int n  = tid & 31;
      const int g  = tid >> 5;            // 0..7
      const int nb = nbrs[p * KNB + n];
      const float dx = pos[nb*3+0] - pos[p*3+0];
      const float dy = pos[nb*3+1] - pos[p*3+1];
      const float dz = pos[nb*3+2] - pos[p*3+2];
      const float r2 = dx*dx + dy*dy;
      const float rr = sqrtf(r2);
      const float inv = (r2 > 0.0f) ? (1.0f / rr) : 0.0f;
      const float c1 = (r2 > 0.0f) ? dx * inv : 1.0f;  // cos(th)
      const float s1 = dy * inv;                        // sin(th)
      const float c2 = c1*c1 - s1*s1;                   // cos(2th)
      const float s2 = 2.0f*c1*s1;                      // sin(2th)

      // T1 columns am in [10g,10g+10): 5 float4 = 5 complex pairs of dc[nbr]
      const float4* dp = reinterpret_cast<const float4*>(data + (size_t)nb*160 + g*20);
      #pragma unroll
      for (int q2 = 0; q2 < 5; ++q2) {
        const float4 v = dp[q2];
        const int am0 = g*10 + q2*2;
        {
          const int mi = (2*q2) % 5;     // ma = mi-2; pha = exp(-i*th*ma)
          const float pr = (mi == 2) ? 1.0f : ((mi == 1 || mi == 3) ? c1 : c2);
          const float pi = (mi == 0) ? s2 : (mi == 1) ? s1 :
                           (mi == 3) ? -s1 : (mi == 4) ? -s2 : 0.0f;
          s.t1r[n][am0] = v.x*pr - v.y*pi;
          s.t1i[n][am0] = v.x*pi + v.y*pr;
        }
        {
          const int mi = (2*q2+1) % 5;
          const float pr = (mi == 2) ? 1.0f : ((mi == 1 || mi == 3) ? c1 : c2);
          const float pi = (mi == 0) ? s2 : (mi == 1) ? s1 :
                           (mi == 3) ? -s1 : (mi == 4) ? -s2 : 0.0f;
          s.t1r[n][am0+1] = v.z*pr - v.w*pi;
          s.t1i[n][am0+1] = v.z*pi + v.w*pr;
        }
      }
      // GP columns fb in [6g,6g+6); fb>=45 padded with zeros
      #pragma unroll
      for (int q = 0; q < 6; ++q) {
        const int fb = g*6 + q;
        float vr = 0.0f, vi = 0.0f;
        if (fb < FBD) {
          const int f    = fb / 5;
          const int b    = fb - 5*f;
          const int irad = f / 3;
          const int iz   = f - 3*irad;
          const float rp  = (float)(irad + 1);      // W0=1, start=1
          const float zp  = (float)(iz - 1);
          const float ddr = rr - rp;
          const float ddz = dz - zp;
          // vol = (4ln2/pi)*(1/(2*pi*rp)) = 0.14046179/rp ; namp = vol^(2/3)/rho
          const float namp = powf(0.14046179f / rp, 0.66666667f) * (1.0f/300.0f);
          const float G = namp * __expf(-2.7725887f * (ddr*ddr + ddz*ddz));
          // phb = exp(+i*th*mb), mb = b-2
          const float pbr = (b == 2) ? 1.0f : ((b == 1 || b == 3) ? c1 : c2);
          const float pbi = (b == 0) ? -s2 : (b == 1) ? -s1 :
                            (b == 3) ?  s1 : (b == 4) ?  s2 : 0.0f;
          vr = G * pbr;
          vi = G * pbi;
        }
        s.gpr[n][fb] = vr;
        s.gpi[n][fb] = vi;
      }
    }
    __syncthreads();

    // ---- GEMM1: U(80x48) = T1^T(80x32) * GP(32x48), complex via 4 real WMMAs
    for (int t = wave; t < 15; t += 8) {          // 5 x 3 tile grid over waves
      const int tm = t / 3;
      const int tn = t - 3*tm;
      v8f p1 = {0,0,0,0,0,0,0,0};
      v8f p2 = {0,0,0,0,0,0,0,0};
      v8f p3 = {0,0,0,0,0,0,0,0};
      v8f p4 = {0,0,0,0,0,0,0,0};
      #pragma unroll
      for (int k = 0; k < 8; ++k) {               // K=32 neighbors, 4 per step
        const int n0 = 4*k + 2*kh;
        v2f ar, ai, br, bi;
        ar.x = s.t1r[n0][tm*16+lc];  ar.y = s.t1r[n0+1][tm*16+lc];
        ai.x = s.t1i[n0][tm*16+lc];  ai.y = s.t1i[n0+1][tm*16+lc];
        br.x = s.gpr[n0][tn*16+lc];  br.y = s.gpr[n0+1][tn*16+lc];
        bi.x = s.gpi[n0][tn*16+lc];  bi.y = s.gpi[n0+1][tn*16+lc];
        p1 = wmma_f32(ar, br, p1);
        p2 = wmma_f32(ai, bi, p2);
        p3 = wmma_f32(ar, bi, p3);
        p4 = wmma_f32(ai, br, p4);
      }
      #pragma unroll
      for (int j = 0; j < 8; ++j) {               // C layout: (M=j+8*kh, N=lc)
        const int am = tm*16 + j + 8*kh;
        const int fb = tn*16 + lc;
        s.ur[pl][am*FBP + fb] = p1[j] - p2[j];    // re = ArBr - AiBi
        s.ui[pl][am*FBP + fb] = p3[j] + p4[j];    // im = ArBi + AiBr
      }
    }
  }
  __syncthreads();

  // ===== GEMM2: per b, Y[16pts x 16ch] = U_b[pts x 800] * W_b[800 x 16] =====
  if (wave < 5) {
    const int b  = wave;
    const int pt = lc & (PT-1);                   // A row = point (dup rows 8..15)
    const float* wb = wts + (size_t)b * 25600;    // this wave's weight slice
    #pragma unroll
    for (int j = 0; j < 4; ++j)                   // warm cache: global_prefetch_b8
      __builtin_prefetch(wb + lane*25 + j*6400, 0, 1);

    v8f q1 = {0,0,0,0,0,0,0,0};
    v8f q2 = {0,0,0,0,0,0,0,0};
    v8f q3 = {0,0,0,0,0,0,0,0};
    v8f q4 = {0,0,0,0,0,0,0,0};

    // main part: K = 720 enumerated as (f outer, am inner; 80 % 4 == 0)
    for (int f = 0; f < 9; ++f) {
      #pragma unroll 4
      for (int ka = 0; ka < 20; ++ka) {
        const int am0 = ka*4 + 2*kh;
        const int u0  = am0*FBP + f*5 + b;
        v2f ar, ai, br, bi;
        ar.x = s.ur[pt][u0];   ar.y = s.ur[pt][u0 + FBP];
        ai.x = s.ui[pt][u0];   ai.y = s.ui[pt][u0 + FBP];
        // W flat: (((b*80+am)*10 + (f+1))*16 + c)*2 + ri ; N lane = c = lc
        const int w0 = (((b*80 + am0)*10 + f + 1)*16 + lc)*2;
        const float2 wv0 = *reinterpret_cast<const float2*>(wts + w0);
        const float2 wv1 = *reinterpret_cast<const float2*>(wts + w0 + 320);
        br.x = wv0.x; bi.x = wv0.y; br.y = wv1.x; bi.y = wv1.y;
        q1 = wmma_f32(ar, br, q1);
        q2 = wmma_f32(ai, bi, q2);
        q3 = wmma_f32(ar, bi, q3);
        q4 = wmma_f32(ai, br, q4);
      }
    }
    // self / identity-filter term (ff = 0), K = 80, A straight from data[p]
    const float* dself = data + (size_t)(pg0 + pt) * 160;
    #pragma unroll 4
    for (int ka = 0; ka < 20; ++ka) {
      const int am0 = ka*4 + 2*kh;
      const float4 dv = *reinterpret_cast<const float4*>(dself + am0*2);
      v2f ar = {dv.x, dv.z};
      v2f ai = {dv.y, dv.w};
      const int w0 = ((b*80 + am0)*10*16 + lc)*2;
      const float2 wv0 = *reinterpret_cast<const float2*>(wts + w0);
      const float2 wv1 = *reinterpret_cast<const float2*>(wts + w0 + 320);
      v2f br = {wv0.x, wv1.x};
      v2f bi = {wv0.y, wv1.y};
      q1 = wmma_f32(ar, br, q1);
      q2 = wmma_f32(ai, bi, q2);
      q3 = wmma_f32(ar, bi, q3);
      q4 = wmma_f32(ai, br, q4);
    }

    if (kh == 0) {  // C rows 0..7 = valid points; N lane = channel c
      const float badd = (b == 2) ? bias[lc] : 0.0f;   // bias on k==0 harmonic, re
      #pragma unroll
      for (int j = 0; j < 8; ++j) {
        const float yr = q1[j] - q2[j] + badd;
        const float yi = q3[j] + q4[j];
        const int pp = pg0 + j;
        const int o  = ((pp*16 + lc)*5 + b)*2;
        out[o]     = yr;
        out[o + 1] = yi;
      }
    }
  }
}

extern "C" void kernel_launch(void* const* d_in, const int* in_sizes, int n_in,
                              void* d_out, int out_size, void* d_ws, size_t ws_size,
                              hipStream_t stream) {
  (void)in_sizes; (void)n_in; (void)d_ws; (void)ws_size; (void)out_size;
  const float* data = (const float*)d_in[0];
  const float* pos  = (const float*)d_in[1];
  const int*   nbr  = (const int*)d_in[2];
  const float* wts  = (const float*)d_in[3];
  const float* bias = (const float*)d_in[4];
  float* out = (float*)d_out;

  // 273 KB dynamic LDS (<= 320 KB/workgroup on CDNA5); set opt-in cap each call
  (void)hipFuncSetAttribute((const void*)PConv3D_kernel,
                            hipFuncAttributeMaxDynamicSharedMemorySize,
                            (int)sizeof(Smem));

  dim3 grid(PNT / PT), block(256);
  hipLaunchKernelGGL(PConv3D_kernel, grid, block, sizeof(Smem), stream,
                     data, pos, nbr, wts, bias, out);
}